// PSELoss_15040975471193
// MI455X (gfx1250) — compile-verified
//
#include <hip/hip_runtime.h>

// ---------------- problem constants (from reference) ----------------
#define NPIX   409600          // 640*640
#define BS     8               // batch
#define NCH    7               // channels (last = full-text map)
#define RATIO_ 3
#define LAMBDA_ 0.7f
#define EPS_    1e-6f

// ---------------- workspace layout ----------------
// [0      .. 65536)  int  hist[8][2048]
// [65536  .. 65568)  int  npos[8]
// [65568  .. 65600)  int  sel_prefix[8]
// [65600  .. 65632)  int  k_rem[8]
// [65632  .. 65664)  int  active[8]
// [65664  .. 65696)  f32  thr[8]
// [65792  .. 108800) f32  partials[8][64][21]
#define WS_HIST 0
#define WS_NPOS 65536
#define WS_SEL  65568
#define WS_KREM 65600
#define WS_ACT  65632
#define WS_THR  65664
#define WS_PART 65792
#define WS_STATE_BYTES 65696

#define NBLK_X 64              // blocks per sample in bulk passes
#define NSUMS  21              // 3 (L_c) + 6*3 (L_s)
#define NVEC   (NPIX / 4)      // 102400 float4 groups per channel-plane

typedef __attribute__((ext_vector_type(2))) float v2f;
typedef __attribute__((ext_vector_type(8))) float v8f;

// One trans-op sigmoid: gfx1250 has V_TANH_F32; sigmoid(x)=0.5+0.5*tanh(x/2).
// Fallback: v_rcp_f32(1+exp(-x)) (2 trans ops, no precise-div sequence).
// Must be used identically in hist + bulk kernels so threshold compares match.
__device__ __forceinline__ float sigmoidf_(float x) {
#if __has_builtin(__builtin_amdgcn_tanhf)
    return fmaf(0.5f, __builtin_amdgcn_tanhf(0.5f * x), 0.5f);
#else
    return __builtin_amdgcn_rcpf(1.0f + __expf(-x));
#endif
}

// ---------------------------------------------------------------
// Radix-select histogram pass over channel 6 (+ mask), per sample.
// round 0: 2048 bins of bits[31:21], also counts n_pos
// round 1: filter bits[31:21]==sel, 2048 bins of bits[20:10]
// round 2: filter bits[31:10]==sel, 1024 bins of bits[9:0]
// ---------------------------------------------------------------
__global__ void hist_kernel(const float* __restrict__ preds,
                            const int* __restrict__ tmask,
                            int* __restrict__ hist,
                            int* __restrict__ npos,
                            const int* __restrict__ selp,
                            const int* __restrict__ active,
                            int round) {
    __shared__ int lh[2048];
    __shared__ int lpos;
    const int b = blockIdx.y;
    if (round > 0 && active[b] == 0) return;     // uniform per block

    for (int i = threadIdx.x; i < 2048; i += blockDim.x) lh[i] = 0;
    if (threadIdx.x == 0) lpos = 0;
    __syncthreads();

    const int sel = (round > 0) ? selp[b] : 0;
    const float4* p6 = (const float4*)(preds + ((size_t)b * NCH + (NCH - 1)) * NPIX);
    const int4*   mk = (const int4*)(tmask + (size_t)b * NPIX);

    int mypos = 0;
    int zcnt  = 0;   // pn == 0.0 -> bin 0, batched to cut LDS-atomic contention
    for (int iv = blockIdx.x * blockDim.x + threadIdx.x; iv < NVEC;
         iv += gridDim.x * blockDim.x) {
        float4 xv = p6[iv];
        int4   mv = mk[iv];
        float xs[4] = {xv.x, xv.y, xv.z, xv.w};
        int   ms[4] = {mv.x, mv.y, mv.z, mv.w};
#pragma unroll
        for (int e = 0; e < 4; ++e) {
            float pnv = sigmoidf_(xs[e]) * (float)ms[e];
            if (pnv >= 0.5f) {
                mypos++;
            } else if (pnv == 0.0f) {
                zcnt++;
            } else {
                unsigned bits = __float_as_uint(pnv);
                if (round == 0) {
                    atomicAdd(&lh[bits >> 21], 1);
                } else if (round == 1) {
                    if ((int)(bits >> 21) == sel) atomicAdd(&lh[(bits >> 10) & 2047], 1);
                } else {
                    if ((int)(bits >> 10) == sel) atomicAdd(&lh[bits & 1023], 1);
                }
            }
        }
    }
    if (zcnt && (round == 0 || sel == 0)) atomicAdd(&lh[0], zcnt);
    if (round == 0 && mypos) atomicAdd(&lpos, mypos);
    __syncthreads();

    for (int i = threadIdx.x; i < 2048; i += blockDim.x) {
        int v = lh[i];
        if (v) atomicAdd(&hist[b * 2048 + i], v);
    }
    if (round == 0 && threadIdx.x == 0 && lpos) atomicAdd(&npos[b], lpos);
}

// ---------------------------------------------------------------
// Scan a sample's histogram top-down to locate the k_rem-th largest,
// update prefix/k_rem, zero the histogram for the next round.
// Launched <<<1,8>>>.
// ---------------------------------------------------------------
__global__ void scan_kernel(int* __restrict__ hist,
                            const int* __restrict__ npos,
                            int* __restrict__ selp,
                            int* __restrict__ krem,
                            int* __restrict__ active,
                            float* __restrict__ thr,
                            int round) {
    const int b = threadIdx.x;
    if (b >= BS) return;
    const int nb        = (round == 2) ? 1024 : 2048;
    const int shiftBits = (round == 2) ? 10 : 11;

    if (round == 0) {
        int np   = npos[b];
        int k    = np * RATIO_;
        int nneg = NPIX - np;
        int act  = (nneg > k) && (k > 0);
        active[b] = act;
        krem[b]   = k;
        selp[b]   = 0;
        thr[b]    = 0.0f;
    }
    if (active[b]) {
        int kr = krem[b];
        int cum = 0, selBin = 0;
        for (int bin = nb - 1; bin >= 0; --bin) {
            int h = hist[b * 2048 + bin];
            cum += h;
            if (cum >= kr) { selBin = bin; kr -= (cum - h); break; }
        }
        krem[b] = kr;
        selp[b] = (selp[b] << shiftBits) | selBin;
        if (round == 2) thr[b] = __uint_as_float((unsigned)selp[b]);
    }
    for (int bin = 0; bin < 2048; ++bin) hist[b * 2048 + bin] = 0;
}

// ---------------------------------------------------------------
// Single bulk pass (B128 loads): all 7 pred channels + 7 label
// channels + mask, 4 pixels per thread-iteration.
//   sums[0..2]        : sum(pn*ln*M), sum(pn^2*M), sum(ln*M)         (L_c)
//   sums[3+3c..3+3c+2]: sum(S*G), sum(S^2), sum(G)  with S=p_c*W,
//                       G=label_c*W (G^2==G)                          (L_s)
// ---------------------------------------------------------------
__global__ void main_sums_kernel(const float* __restrict__ preds,
                                 const int* __restrict__ labels,
                                 const int* __restrict__ tmask,
                                 const int* __restrict__ active,
                                 const float* __restrict__ thr,
                                 float* __restrict__ partials) {
    const int b   = blockIdx.y;
    const int tid = threadIdx.x;
    const bool act = active[b] != 0;
    const float t  = thr[b];

    const float4* p6 = (const float4*)(preds  + ((size_t)b * NCH + (NCH - 1)) * NPIX);
    const int4*   l6 = (const int4*)(labels + ((size_t)b * NCH + (NCH - 1)) * NPIX);
    const int4*   mk = (const int4*)(tmask  + (size_t)b * NPIX);

    float acc[NSUMS];
#pragma unroll
    for (int j = 0; j < NSUMS; ++j) acc[j] = 0.0f;

    const int vstride = NBLK_X * 256;            // stride in float4 groups
    for (int iv = blockIdx.x * 256 + tid; iv < NVEC; iv += vstride) {
        if (iv + vstride < NVEC) {               // streaming prefetch hints
            __builtin_prefetch(&p6[iv + vstride], 0, 3);
            __builtin_prefetch(&l6[iv + vstride], 0, 3);
        }
        int4   mv  = mk[iv];
        float4 x6v = p6[iv];
        int4   l6v = l6[iv];

        float xs[4] = {x6v.x, x6v.y, x6v.z, x6v.w};
        int   ms[4] = {mv.x, mv.y, mv.z, mv.w};
        int   ls[4] = {l6v.x, l6v.y, l6v.z, l6v.w};

        float W[4], Mv[4];
#pragma unroll
        for (int e = 0; e < 4; ++e) {
            float m   = (float)ms[e];
            float pnv = sigmoidf_(xs[e]) * m;
            float ln  = (float)ls[e] * m;
            W[e]  = (pnv >= 0.5f) ? 1.0f : 0.0f;
            Mv[e] = act ? ((pnv >= t) ? 1.0f : 0.0f) : 1.0f;
            acc[0] += pnv * ln * Mv[e];     // S*G  (M^2 == M)
            acc[1] += pnv * pnv * Mv[e];    // S^2
            acc[2] += ln * Mv[e];           // G^2 == G (ln in {0,1})
        }

#pragma unroll
        for (int c = 0; c < NCH - 1; ++c) {
            const float4* pc = (const float4*)(preds  + ((size_t)b * NCH + c) * NPIX);
            const int4*   lc = (const int4*)(labels + ((size_t)b * NCH + c) * NPIX);
            float4 xv = pc[iv];
            int4   lv = lc[iv];
            float xc[4] = {xv.x, xv.y, xv.z, xv.w};
            int   lcb[4] = {lv.x, lv.y, lv.z, lv.w};
#pragma unroll
            for (int e = 0; e < 4; ++e) {
                float S = sigmoidf_(xc[e]) * W[e];
                float G = (float)lcb[e] * W[e];   // in {0,1}
                acc[3 + 3 * c + 0] += S * G;
                acc[3 + 3 * c + 1] += S * S;
                acc[3 + 3 * c + 2] += G;          // G^2 == G
            }
        }
    }

    // wave32 tree reduction (fixed order -> deterministic)
#pragma unroll
    for (int j = 0; j < NSUMS; ++j) {
        float v = acc[j];
        for (int off = 16; off >= 1; off >>= 1) v += __shfl_down(v, off, 32);
        acc[j] = v;
    }

    __shared__ float ws[8][NSUMS];   // 8 waves per 256-thread block
    const int lane = tid & 31, wave = tid >> 5;
    if (lane == 0) {
#pragma unroll
        for (int j = 0; j < NSUMS; ++j) ws[wave][j] = acc[j];
    }
    __syncthreads();
    if (tid < NSUMS) {
        float s = 0.0f;
        for (int w = 0; w < 8; ++w) s += ws[w][tid];
        partials[((size_t)b * NBLK_X + blockIdx.x) * NSUMS + tid] = s;
    }
}

// ---------------------------------------------------------------
// Finalize: reduce 64 block-partials per (sample, sum) with
// V_WMMA_F32_16X16X4_F32 (A = 16x4 = exactly 64 f32, B = ones),
// computing each dice group on the fly (no sums[] spill).
// 1 block, 8 waves <-> 8 samples.
// ---------------------------------------------------------------
__global__ void finalize_kernel(const float* __restrict__ partials,
                                float* __restrict__ out) {
    __shared__ float res[BS][3];
    const int tid  = threadIdx.x;
    const int lane = tid & 31;
    const int b    = tid >> 5;

    // A-matrix 16x4 f32 layout: lanes 0-15 hold K=0,1; lanes 16-31 hold K=2,3; M = lane%16
    const int m    = lane & 15;
    const int kb   = (lane < 16) ? 0 : 2;
    const float* pp = partials + (size_t)b * NBLK_X * NSUMS;

    v2f ones; ones.x = 1.0f; ones.y = 1.0f;

    float Lc = 0.0f, dsum = 0.0f;
    for (int g = 0; g < 7; ++g) {              // group 0 = L_c, 1..6 = L_s channels
        float s3[3];
#pragma unroll
        for (int q = 0; q < 3; ++q) {
            const int j = g * 3 + q;
            v2f a;
            a.x = pp[(size_t)(m * 4 + kb + 0) * NSUMS + j];
            a.y = pp[(size_t)(m * 4 + kb + 1) * NSUMS + j];
            v8f c = {};
            // D[m][n] = rowsum(A[m][:]) since B == ones
            v8f d = __builtin_amdgcn_wmma_f32_16x16x4_f32(
                false, a, false, ones, (short)0, c, false, false);
            float s = d[0] + d[1] + d[2] + d[3] + d[4] + d[5] + d[6] + d[7];
            s += __shfl_xor(s, 16, 32);        // lanes 0-15: M 0-7, lanes 16-31: M 8-15
            s3[q] = s;
        }
        float dice = 2.0f * s3[0] / (s3[1] + s3[2] + EPS_);
        if (g == 0) Lc = 1.0f - dice;
        else        dsum += dice;
    }
    float Ls = 1.0f - dsum / (float)(NCH - 1);

    if (lane == 0) {
        res[b][0] = Lc;
        res[b][1] = Ls;
        res[b][2] = LAMBDA_ * Lc + (1.0f - LAMBDA_) * Ls;
    }
    __syncthreads();
    if (tid < 3) {
        float s = 0.0f;
        for (int bb = 0; bb < BS; ++bb) s += res[bb][tid];
        out[tid] = s / (float)BS;
    }
}

// ---------------------------------------------------------------
extern "C" void kernel_launch(void* const* d_in, const int* in_sizes, int n_in,
                              void* d_out, int out_size, void* d_ws, size_t ws_size,
                              hipStream_t stream) {
    const float* preds  = (const float*)d_in[0];
    const int*   labels = (const int*)d_in[1];
    const int*   tmask  = (const int*)d_in[2];
    float*       out    = (float*)d_out;

    char* ws = (char*)d_ws;
    int*   hist = (int*)(ws + WS_HIST);
    int*   npos = (int*)(ws + WS_NPOS);
    int*   selp = (int*)(ws + WS_SEL);
    int*   krem = (int*)(ws + WS_KREM);
    int*   act  = (int*)(ws + WS_ACT);
    float* thr  = (float*)(ws + WS_THR);
    float* part = (float*)(ws + WS_PART);

    // re-init select/histogram state every call (ws is poisoned, not re-zeroed)
    hipMemsetAsync(d_ws, 0, WS_STATE_BYTES, stream);

    dim3 hg(NBLK_X, BS);
    hist_kernel<<<hg, 256, 0, stream>>>(preds, tmask, hist, npos, selp, act, 0);
    scan_kernel<<<1, BS, 0, stream>>>(hist, npos, selp, krem, act, thr, 0);
    hist_kernel<<<hg, 256, 0, stream>>>(preds, tmask, hist, npos, selp, act, 1);
    scan_kernel<<<1, BS, 0, stream>>>(hist, npos, selp, krem, act, thr, 1);
    hist_kernel<<<hg, 256, 0, stream>>>(preds, tmask, hist, npos, selp, act, 2);
    scan_kernel<<<1, BS, 0, stream>>>(hist, npos, selp, krem, act, thr, 2);

    main_sums_kernel<<<hg, 256, 0, stream>>>(preds, labels, tmask, act, thr, part);
    finalize_kernel<<<1, 256, 0, stream>>>(part, out);
}